// STPGCNs_40931038331184
// MI455X (gfx1250) — compile-verified
//
#include <hip/hip_runtime.h>
#include <math.h>

typedef float v2f __attribute__((ext_vector_type(2)));
typedef float v8f __attribute__((ext_vector_type(8)));

#define B_ 4
#define T_ 12
#define V_ 512
#define C_ 64
#define D_ 8       // pos-emb dim d
#define TS_ 3
#define P_ 12
#define L_ 4
#define C2_ 128    // 2*C
#define KW_ (TS_*V_)  // 1536

__device__ __forceinline__ float sigm(float x) { return 1.0f / (1.0f + expf(-x)); }

// -------------------- input layer: h = x*Wi + bi  (D=1) --------------------
__global__ void k_input(const float* __restrict__ x, const float* __restrict__ Wi,
                        const float* __restrict__ bi, float* __restrict__ h) {
  int idx = blockIdx.x * blockDim.x + threadIdx.x;
  if (idx >= B_ * T_ * V_ * C_) return;
  int c = idx & (C_ - 1);
  int rest = idx >> 6;               // (b*T+t)*V+v
  h[idx] = x[rest] * Wi[c] + bi[c];
}

// -------------------- per-layer small gaussians ----------------------------
// gc0[i]=gc(sape,0)[i], gc1[j]=gc(sape,1)[j], ept[b,t,k]=exp(per_t[b,t,k])
__global__ void k_small(const float* __restrict__ sape, const float* __restrict__ tape,
                        const float* __restrict__ trpe, const float* __restrict__ mu,
                        const float* __restrict__ sig, float* __restrict__ gc0,
                        float* __restrict__ gc1, float* __restrict__ ept) {
  int tid = threadIdx.x;
  if (tid < V_) {
    const float* e = sape + tid * D_;
    float s0 = 0.f, s1 = 0.f;
#pragma unroll
    for (int dd = 0; dd < D_; ++dd) {
      float d0 = e[dd] - mu[dd];        float g0 = sig[dd];
      float d1 = e[dd] - mu[D_ + dd];   float g1 = sig[D_ + dd];
      s0 -= 0.5f * d0 * d0 * g0 * g0;
      s1 -= 0.5f * d1 * d1 * g1 * g1;
    }
    gc0[tid] = s0;
    gc1[tid] = s1;
  } else if (tid < V_ + B_ * T_ * TS_) {
    int q = tid - V_;
    int b = q / (T_ * TS_);
    int r = q % (T_ * TS_);
    int t = r / TS_;
    int k = r % TS_;
    float s = 0.f;
#pragma unroll
    for (int dd = 0; dd < D_; ++dd) {       // gc(tape,2)[b,t]
      float x2 = tape[(b * T_ + t) * D_ + dd] - mu[2 * D_ + dd];
      float g2 = sig[2 * D_ + dd];
      s -= 0.5f * x2 * x2 * g2 * g2;
    }
    int tt = t + k;                          // gc(tape_pad,3)[b,t+k]; pad rows are ZERO embeddings
#pragma unroll
    for (int dd = 0; dd < D_; ++dd) {
      float e3 = (tt < 2) ? 0.f : tape[(b * T_ + tt - 2) * D_ + dd];
      float x3 = e3 - mu[3 * D_ + dd];
      float g3 = sig[3 * D_ + dd];
      s -= 0.5f * x3 * x3 * g3 * g3;
    }
#pragma unroll
    for (int dd = 0; dd < D_; ++dd) {       // gc(trpe,5)[k]
      float x5 = trpe[k * D_ + dd] - mu[5 * D_ + dd];
      float g5 = sig[5 * D_ + dd];
      s -= 0.5f * x5 * x5 * g5 * g5;
    }
    ept[q] = expf(s);
  }
}

// -------------------- EbT[j*V+i] = exp(gc0[i]+gc1[j]+gc_srpe(i,j)) ---------
__global__ void k_ebt(const float* __restrict__ srpe, const float* __restrict__ gc0,
                      const float* __restrict__ gc1, const float* __restrict__ mu,
                      const float* __restrict__ sig, float* __restrict__ EbT) {
  int idx = blockIdx.x * blockDim.x + threadIdx.x;  // 262144 threads
  int j = idx >> 9;
  int i = idx & (V_ - 1);
  const float* e = srpe + (i * V_ + j) * D_;        // srpe is (V_i, V_j, d)
  float s = gc0[i] + gc1[j];
#pragma unroll
  for (int dd = 0; dd < D_; ++dd) {
    float x = e[dd] - mu[4 * D_ + dd];
    float g = sig[4 * D_ + dd];
    s -= 0.5f * x * x * g * g;
  }
  EbT[idx] = expf(s);
}

// -------------------- main STPGConv: fp32 WMMA ------------------------------
// Wave computes g[16(j) x 64(c)] = sum_k ept * ((EbT.*mask) @ h_pad), then
// through LDS, h_new = GLU( g @ Wg + bg + sape@Wsp + tape@Wtp ).
__global__ void __launch_bounds__(128)
k_stpg(const float* __restrict__ hin, const float* __restrict__ EbT,
       const float* __restrict__ mask, const float* __restrict__ ept,
       const float* __restrict__ Wg, const float* __restrict__ bg,
       const float* __restrict__ sape, const float* __restrict__ Wsp,
       const float* __restrict__ tape, const float* __restrict__ Wtp,
       float* __restrict__ hout) {
  __shared__ float smem[4 * 16 * 65];   // per-wave 16x64 tile, stride 65 vs bank conflicts
  const int lane  = threadIdx.x & 31;
  const int wrp   = threadIdx.x >> 5;
  const int jt    = blockIdx.x * 4 + wrp;
  const int t     = blockIdx.y;
  const int b     = blockIdx.z;
  const int j0    = jt * 16;
  const int mrow  = lane & 15;     // A: M row / B,C,D: N column (within 16)
  const int khalf = lane >> 4;     // 0: K{0,1} or M{0..7}; 1: K{2,3} or M{8..15}

  const v8f vzero = {0.f, 0.f, 0.f, 0.f, 0.f, 0.f, 0.f, 0.f};
  v8f acc[4];
#pragma unroll
  for (int ct = 0; ct < 4; ++ct) acc[ct] = vzero;

  // ---- stage 1: g = sum_k ept[b,t,k] * (EbT[j,:].*mask[b,j,kV+:]) @ h[b,t+k-2]
  const float* eb = EbT + (j0 + mrow) * V_;
  for (int kk = 0; kk < TS_; ++kk) {
    int tp = t + kk - 2;                      // x_pad zero rows -> skip (uniform branch)
    if (tp < 0) continue;
    float sc = ept[(b * T_ + t) * TS_ + kk];
    const float* hsl = hin + (size_t)(b * T_ + tp) * V_ * C_;
    const float* mk  = mask + (size_t)(b * V_ + j0 + mrow) * KW_ + kk * V_;
    const float* hb  = hsl + khalf * 2 * C_ + mrow;
    for (int i0 = 0; i0 < V_; i0 += 4) {
      int acol = i0 + khalf * 2;
      v2f a;                                   // A[16x4]: lane=m+16*(k/2), vgpr=k%2
      a.x = eb[acol]     * mk[acol]     * sc;
      a.y = eb[acol + 1] * mk[acol + 1] * sc;
      const float* hp = hb + i0 * C_;
#pragma unroll
      for (int ct = 0; ct < 4; ++ct) {
        v2f bb;                                // B[4x16]: lane=n+16*(k/2), vgpr=k%2
        bb.x = hp[ct * 16];
        bb.y = hp[ct * 16 + C_];
        acc[ct] = __builtin_amdgcn_wmma_f32_16x16x4_f32(
            false, a, false, bb, (short)0, acc[ct], false, false);
      }
    }
  }

  // ---- spill g tile to LDS (re-layout C->A) ----
  float* sm = smem + wrp * 16 * 65;
#pragma unroll
  for (int ct = 0; ct < 4; ++ct)
#pragma unroll
    for (int r = 0; r < 8; ++r)
      sm[(r + 8 * khalf) * 65 + ct * 16 + mrow] = acc[ct][r];
  __syncthreads();

  // ---- stage 2: z = g @ Wg + bg + sape@Wsp + tape@Wtp ; accumulator holds biases
  float tp8[8];
#pragma unroll
  for (int dd = 0; dd < D_; ++dd) tp8[dd] = tape[(b * T_ + t) * D_ + dd];
  v8f acc2[8];
#pragma unroll
  for (int ct = 0; ct < 8; ++ct) {
    int n = ct * 16 + mrow;
    float te = bg[n];
#pragma unroll
    for (int dd = 0; dd < D_; ++dd) te += tp8[dd] * Wtp[dd * C2_ + n];
#pragma unroll
    for (int r = 0; r < 8; ++r) {
      int j = j0 + r + 8 * khalf;
      float se = 0.f;
#pragma unroll
      for (int dd = 0; dd < D_; ++dd) se += sape[j * D_ + dd] * Wsp[dd * C2_ + n];
      acc2[ct][r] = te + se;
    }
  }
  for (int i0 = 0; i0 < C_; i0 += 4) {
    int kb = i0 + khalf * 2;
    v2f a;
    a.x = sm[mrow * 65 + kb];
    a.y = sm[mrow * 65 + kb + 1];
#pragma unroll
    for (int ct = 0; ct < 8; ++ct) {
      v2f bb;
      bb.x = Wg[kb * C2_ + ct * 16 + mrow];
      bb.y = Wg[(kb + 1) * C2_ + ct * 16 + mrow];
      acc2[ct] = __builtin_amdgcn_wmma_f32_16x16x4_f32(
          false, a, false, bb, (short)0, acc2[ct], false, false);
    }
  }

  // ---- GLU: h_new[:,c] = z[:,c] * sigmoid(z[:,c+64]) ----
#pragma unroll
  for (int ct = 0; ct < 4; ++ct)
#pragma unroll
    for (int r = 0; r < 8; ++r) {
      float zl = acc2[ct][r];
      float zr = acc2[ct + 4][r];
      int m = r + 8 * khalf;
      int c = ct * 16 + mrow;
      hout[((size_t)(b * T_ + t) * V_ + j0 + m) * C_ + c] = zl * sigm(zr);
    }
}

// -------------------- GFS: y[b,o,v] = sum_{c,t} h[b,t,v,c]*fs_w[o,c,t]+fs_b[o]
__global__ void k_gfs(const float* __restrict__ h, const float* __restrict__ fsw,
                      const float* __restrict__ fsb, float* __restrict__ y) {
  int idx = blockIdx.x * blockDim.x + threadIdx.x;  // B*64*V
  int v = idx & (V_ - 1);
  int o = (idx >> 9) & 63;
  int b = idx >> 15;
  float acc = fsb[o];
  for (int c = 0; c < C_; ++c) {
    const float* fw = fsw + (o * C_ + c) * T_;
#pragma unroll
    for (int t = 0; t < T_; ++t)
      acc += h[((size_t)(b * T_ + t) * V_ + v) * C_ + c] * fw[t];
  }
  y[idx] = acc;   // layout (b,o,v)
}

// -------------------- GLU1 -> skip slice in scat ---------------------------
__global__ void k_glu1(const float* __restrict__ y, const float* __restrict__ w,
                       const float* __restrict__ bb, float* __restrict__ scat, int l) {
  int idx = blockIdx.x * blockDim.x + threadIdx.x;  // B*64*V
  int v = idx & (V_ - 1);
  int o = (idx >> 9) & 63;
  int b = idx >> 15;
  const float* yb = y + (size_t)b * C_ * V_ + v;
  float z1 = bb[o], z2 = bb[o + C_];
  for (int c = 0; c < C_; ++c) {
    float yv = yb[c * V_];
    z1 += w[o * C_ + c] * yv;
    z2 += w[(o + C_) * C_ + c] * yv;
  }
  scat[((size_t)b * 4 * C_ + l * C_ + o) * V_ + v] = z1 * sigm(z2);
}

// -------------------- output GLU over concatenated skips -------------------
__global__ void k_gout(const float* __restrict__ scat, const float* __restrict__ w,
                       const float* __restrict__ bb, float* __restrict__ sfin) {
  int idx = blockIdx.x * blockDim.x + threadIdx.x;  // B*256*V
  int v = idx & (V_ - 1);
  int o = (idx >> 9) & 255;
  int b = idx >> 17;
  const float* sc = scat + (size_t)b * 256 * V_ + v;
  float z1 = bb[o], z2 = bb[o + 256];
  for (int c = 0; c < 256; ++c) {
    float s = sc[c * V_];
    z1 += w[o * 256 + c] * s;
    z2 += w[(o + 256) * 256 + c] * s;
  }
  sfin[((size_t)b * 256 + o) * V_ + v] = z1 * sigm(z2);
}

// -------------------- final projection out_w @ s + out_b -------------------
__global__ void k_out(const float* __restrict__ sfin, const float* __restrict__ w,
                      const float* __restrict__ bb, float* __restrict__ out) {
  int idx = blockIdx.x * blockDim.x + threadIdx.x;  // B*P*V
  if (idx >= B_ * P_ * V_) return;
  int v = idx & (V_ - 1);
  int p = (idx >> 9) % P_;
  int b = idx / (P_ * V_);
  const float* sf = sfin + (size_t)b * 256 * V_ + v;
  float acc = bb[p];
  for (int c = 0; c < 256; ++c) acc += w[p * 256 + c] * sf[c * V_];
  out[idx] = acc;
}

// ===========================================================================
extern "C" void kernel_launch(void* const* d_in, const int* in_sizes, int n_in,
                              void* d_out, int out_size, void* d_ws, size_t ws_size,
                              hipStream_t stream) {
  (void)in_sizes; (void)n_in; (void)out_size; (void)ws_size;
  const float* x     = (const float*)d_in[0];
  const float* sape  = (const float*)d_in[1];
  const float* tape  = (const float*)d_in[2];
  const float* srpe  = (const float*)d_in[3];
  const float* trpe  = (const float*)d_in[4];
  // d_in[5] zeros_x, d_in[6] zeros_tape: known-zero, handled analytically
  const float* mask  = (const float*)d_in[7];
  const float* Wi    = (const float*)d_in[8];
  const float* bi    = (const float*)d_in[9];
  const float* gc_mu = (const float*)d_in[10];
  const float* gc_sig= (const float*)d_in[11];
  const float* Wg    = (const float*)d_in[12];
  const float* bg    = (const float*)d_in[13];
  const float* Wsp   = (const float*)d_in[14];
  const float* Wtp   = (const float*)d_in[15];
  const float* fs_w  = (const float*)d_in[16];
  const float* fs_b  = (const float*)d_in[17];
  const float* g1w   = (const float*)d_in[18];
  const float* g1b   = (const float*)d_in[19];
  const float* gow   = (const float*)d_in[20];
  const float* gob   = (const float*)d_in[21];
  const float* outw  = (const float*)d_in[22];
  const float* outb  = (const float*)d_in[23];
  float* out = (float*)d_out;

  float* ws = (float*)d_ws;
  float* hA   = ws;  ws += (size_t)B_ * T_ * V_ * C_;   // 1572864
  float* hB   = ws;  ws += (size_t)B_ * T_ * V_ * C_;   // 1572864
  float* EbT  = ws;  ws += (size_t)V_ * V_;             // 262144
  float* gc0  = ws;  ws += V_;
  float* gc1  = ws;  ws += V_;
  float* eptb = ws;  ws += 256;                          // 144 used
  float* ybuf = ws;  ws += (size_t)B_ * C_ * V_;        // 131072
  float* scat = ws;  ws += (size_t)B_ * 4 * C_ * V_;    // 524288
  float* sfin = ws;  ws += (size_t)B_ * 4 * C_ * V_;    // 524288

  k_input<<<(B_ * T_ * V_ * C_ + 255) / 256, 256, 0, stream>>>(x, Wi, bi, hA);

  float* hin = hA;
  float* hout = hB;
  for (int l = 0; l < L_; ++l) {
    const float* mu = gc_mu + l * 6 * D_;
    const float* sg = gc_sig + l * 6 * D_;
    k_small<<<1, 704, 0, stream>>>(sape, tape, trpe, mu, sg, gc0, gc1, eptb);
    k_ebt<<<(V_ * V_) / 256, 256, 0, stream>>>(srpe, gc0, gc1, mu, sg, EbT);
    k_stpg<<<dim3(V_ / 16 / 4, T_, B_), 128, 0, stream>>>(
        hin, EbT, mask, eptb,
        Wg + (size_t)l * C_ * C2_, bg + l * C2_,
        sape, Wsp + (size_t)l * D_ * C2_,
        tape, Wtp + (size_t)l * D_ * C2_, hout);
    k_gfs<<<(B_ * C_ * V_) / 256, 256, 0, stream>>>(
        hout, fs_w + (size_t)l * C_ * C_ * T_, fs_b + l * C_, ybuf);
    k_glu1<<<(B_ * C_ * V_) / 256, 256, 0, stream>>>(
        ybuf, g1w + (size_t)l * C2_ * C_, g1b + l * C2_, scat, l);
    float* tmp = hin; hin = hout; hout = tmp;
  }

  k_gout<<<(B_ * 256 * V_) / 256, 256, 0, stream>>>(scat, gow, gob, sfin);
  k_out<<<(B_ * P_ * V_ + 255) / 256, 256, 0, stream>>>(sfin, outw, outb, out);
}